// ConcatNonLocalBlock_36103495090208
// MI455X (gfx1250) — compile-verified
//
#include <hip/hip_runtime.h>

typedef __attribute__((ext_vector_type(2))) float v2f;
typedef __attribute__((ext_vector_type(8))) float v8f;

#define BB 8
#define CC 256
#define EE 128
#define NN 3136
#define NT 64
#define NTILES (NN / NT) /* 49 */

// ---------------------------------------------------------------------------
// Kernel 1: fold Wcat into Wq/Wk:  a[c] = sum_e wq_c[e]*Wq[e,c] + wk_c[e]*Wk[e,c]
//           a0 = wq_c . bq + wk_c . bk    (stored at a_out[CC])
// ---------------------------------------------------------------------------
__global__ __launch_bounds__(256) void prep_a_kernel(
    const float* __restrict__ Wq, const float* __restrict__ bq,
    const float* __restrict__ Wk, const float* __restrict__ bk,
    const float* __restrict__ Wcat, float* __restrict__ a_out) {
  __shared__ float red[128];
  const int c = threadIdx.x;  // 0..255
  float acc = 0.f;
  for (int e = 0; e < EE; ++e)
    acc = fmaf(Wcat[e], Wq[e * CC + c], fmaf(Wcat[EE + e], Wk[e * CC + c], acc));
  a_out[c] = acc;
  if (c < EE) red[c] = Wcat[c] * bq[c] + Wcat[EE + c] * bk[c];
  __syncthreads();
  for (int s = 64; s > 0; s >>= 1) {
    if (c < s) red[c] += red[c + s];
    __syncthreads();
  }
  if (c == 0) a_out[CC] = red[0];
}

// ---------------------------------------------------------------------------
// Kernel 2: xsum[b,c] = sum_n x[b,c,n]   (one block per (b,c) row)
// ---------------------------------------------------------------------------
__global__ __launch_bounds__(256) void row_sum_kernel(
    const float* __restrict__ x, float* __restrict__ xsum) {
  const int row = blockIdx.x;  // b*CC + c
  const float4* __restrict__ p =
      reinterpret_cast<const float4*>(x + (size_t)row * NN);
  float acc = 0.f;
  for (int i = threadIdx.x; i < NN / 4; i += 256) {
    float4 v = p[i];
    acc += v.x + v.y + v.z + v.w;
  }
  __shared__ float red[256];
  red[threadIdx.x] = acc;
  __syncthreads();
  for (int s = 128; s > 0; s >>= 1) {
    if (threadIdx.x < s) red[threadIdx.x] += red[threadIdx.x + s];
    __syncthreads();
  }
  if (threadIdx.x == 0) xsum[row] = red[0];
}

// ---------------------------------------------------------------------------
// Kernel 3: vsum[b,e] = Wv . xsum[b] + n*bv[e];  g[b,c] = (Wexp . vsum[b])[c]/n
// ---------------------------------------------------------------------------
__global__ __launch_bounds__(256) void prep_g_kernel(
    const float* __restrict__ Wv, const float* __restrict__ bv,
    const float* __restrict__ Wexp, const float* __restrict__ xsum,
    float* __restrict__ g) {
  __shared__ float xs[BB * CC];  // 2048
  __shared__ float vs[BB * EE];  // 1024
  for (int i = threadIdx.x; i < BB * CC; i += 256) xs[i] = xsum[i];
  __syncthreads();
  for (int i = threadIdx.x; i < BB * EE; i += 256) {
    const int b = i >> 7, e = i & (EE - 1);
    float acc = (float)NN * bv[e];
    const float* __restrict__ wr = Wv + e * CC;
    const float* __restrict__ xr = xs + b * CC;
    for (int c = 0; c < CC; ++c) acc = fmaf(wr[c], xr[c], acc);
    vs[i] = acc;
  }
  __syncthreads();
  const float inv_n = 1.0f / (float)NN;
  for (int i = threadIdx.x; i < BB * CC; i += 256) {
    const int b = i >> 8, c = i & (CC - 1);
    float acc = 0.f;
    const float* __restrict__ wr = Wexp + c * EE;
    const float* __restrict__ vr = vs + b * EE;
    for (int e = 0; e < EE; ++e) acc = fmaf(wr[e], vr[e], acc);
    g[i] = acc * inv_n;
  }
}

// ---------------------------------------------------------------------------
// Kernel 4: fused main pass over x.
//   stage 0: LDS <- x[b, 0:256, n0:n0+64]
//   stage 1: s[n] = relu( a . x[:, n] + a0 )  via V_WMMA_F32_16X16X4_F32
//            D = A * B with A = X^T tile (16 n x 4 c), B = a[c0..c0+3]
//            broadcast across all 16 columns -> D[:,0] holds partial s.
//            8 waves = 4 n-subtiles x 2 c-halves (K accumulated 32 steps).
//   stage 2: out[b,c,n] = s[n]*g[b,c] + bexp[c] + x[b,c,n]  (float4 stores)
// ---------------------------------------------------------------------------
__global__ __launch_bounds__(256) void fused_out_kernel(
    const float* __restrict__ x, const float* __restrict__ a_vec,
    const float* __restrict__ g, const float* __restrict__ bexp,
    float* __restrict__ out) {
  __shared__ float xt[CC * NT];    // 64 KB, row-major [c][n]
  __shared__ float spart[2][NT];   // partial s per c-half
  __shared__ float s_arr[NT];

  const int b = blockIdx.y;
  const int n0 = blockIdx.x * NT;
  const int tid = threadIdx.x;
  const float a0 = a_vec[CC];  // uniform -> scalar load

  // ---- stage 0: tile load, 16 float4 per thread, fully coalesced ----
#pragma unroll
  for (int i = 0; i < 16; ++i) {
    const int f = i * 256 + tid;  // 0..4095 float4 slots
    const int c = f >> 4;
    const int j4 = f & 15;
    const float4 v = *reinterpret_cast<const float4*>(
        x + ((size_t)(b * CC + c)) * NN + n0 + j4 * 4);
    *reinterpret_cast<float4*>(&xt[c * NT + j4 * 4]) = v;
  }
  __syncthreads();

  // ---- stage 1: WMMA f32 16x16x4, all 8 waves active (EXEC all ones) ----
  {
    const int wave = tid >> 5;
    const int lane = tid & 31;
    const int st = wave & 3;        // n-subtile (16 n values)
    const int h = wave >> 1 >> 1;   // c-half (0: c 0..127, 1: c 128..255)
    const int nidx = st * 16 + (lane & 15);
    const int koff = (lane < 16) ? 0 : 2;  // A lanes 0-15: K=0,1; 16-31: K=2,3
    v8f acc = {};
    for (int kk = 0; kk < 32; ++kk) {
      const int cb = h * 128 + kk * 4;  // K-block base channel
      const int c0 = cb + koff;
      v2f A;
      A.x = xt[c0 * NT + nidx];        // A[m, K]   (K = koff)
      A.y = xt[(c0 + 1) * NT + nidx];  // A[m, K+1]
      v2f Bv;
      Bv.x = a_vec[c0];                // B[K,   n] broadcast over n
      Bv.y = a_vec[c0 + 1];            // B[K+1, n] broadcast over n
      acc = __builtin_amdgcn_wmma_f32_16x16x4_f32(
          false, A, false, Bv, (short)0, acc, false, false);
    }
    // D column 0: lane 0 holds M=0..7 (vgpr 0..7), lane 16 holds M=8..15
    if (lane == 0) {
#pragma unroll
      for (int r = 0; r < 8; ++r) spart[h][st * 16 + r] = acc[r];
    } else if (lane == 16) {
#pragma unroll
      for (int r = 0; r < 8; ++r) spart[h][st * 16 + 8 + r] = acc[r];
    }
  }
  __syncthreads();
  if (tid < NT) {
    const float sv = spart[0][tid] + spart[1][tid] + a0;
    s_arr[tid] = sv > 0.f ? sv : 0.f;
  }
  __syncthreads();

  // ---- stage 2: epilogue, float4 coalesced stores ----
#pragma unroll
  for (int i = 0; i < 16; ++i) {
    const int f = i * 256 + tid;
    const int c = f >> 4;
    const int j4 = f & 15;
    const float gv = g[b * CC + c];
    const float be = bexp[c];
    const float4 xv = *reinterpret_cast<const float4*>(&xt[c * NT + j4 * 4]);
    float4 o;
    o.x = fmaf(s_arr[j4 * 4 + 0], gv, be + xv.x);
    o.y = fmaf(s_arr[j4 * 4 + 1], gv, be + xv.y);
    o.z = fmaf(s_arr[j4 * 4 + 2], gv, be + xv.z);
    o.w = fmaf(s_arr[j4 * 4 + 3], gv, be + xv.w);
    *reinterpret_cast<float4*>(out + ((size_t)(b * CC + c)) * NN + n0 + j4 * 4) = o;
  }
}

// ---------------------------------------------------------------------------
extern "C" void kernel_launch(void* const* d_in, const int* in_sizes, int n_in,
                              void* d_out, int out_size, void* d_ws,
                              size_t ws_size, hipStream_t stream) {
  const float* x    = (const float*)d_in[0];
  const float* Wq   = (const float*)d_in[1];
  const float* bq   = (const float*)d_in[2];
  const float* Wk   = (const float*)d_in[3];
  const float* bk   = (const float*)d_in[4];
  const float* Wv   = (const float*)d_in[5];
  const float* bv   = (const float*)d_in[6];
  const float* Wcat = (const float*)d_in[7];
  const float* Wexp = (const float*)d_in[8];
  const float* bexp = (const float*)d_in[9];
  float* out = (float*)d_out;

  float* ws_f = (float*)d_ws;
  float* a_vec = ws_f;              // 257 floats (a[256] + a0), padded to 320
  float* xsum = ws_f + 320;         // 2048 floats
  float* gmat = ws_f + 320 + 2048;  // 2048 floats

  prep_a_kernel<<<1, 256, 0, stream>>>(Wq, bq, Wk, bk, Wcat, a_vec);
  row_sum_kernel<<<BB * CC, 256, 0, stream>>>(x, xsum);
  prep_g_kernel<<<1, 256, 0, stream>>>(Wv, bv, Wexp, xsum, gmat);
  fused_out_kernel<<<dim3(NTILES, BB), 256, 0, stream>>>(x, a_vec, gmat, bexp,
                                                         out);
}